// SimpleGCNAF_9474697855477
// MI455X (gfx1250) — compile-verified
//
#include <hip/hip_runtime.h>
#include <math.h>

typedef __attribute__((ext_vector_type(2))) float v2f;
typedef __attribute__((ext_vector_type(8))) float v8f;

#define D_FEAT 128
#define C_OUT  40
#define C_PAD  48   // 3 WMMA n-tiles of 16

// ---------------- utility ----------------
__global__ void zero_f32(float* __restrict__ p, int n) {
    int i = blockIdx.x * blockDim.x + threadIdx.x;
    if (i < n) p[i] = 0.0f;
}

// Wpad[c][d] = (c < 40) ? lin[d][c] : 0   -- transposed + zero-padded weights
__global__ void pad_weights_kernel(const float* __restrict__ lin,
                                   float* __restrict__ Wpad) {
    int i = blockIdx.x * blockDim.x + threadIdx.x;   // 0 .. 48*128-1
    if (i >= C_PAD * D_FEAT) return;
    int c = i / D_FEAT;
    int d = i - c * D_FEAT;
    Wpad[i] = (c < C_OUT) ? lin[d * C_OUT + c] : 0.0f;
}

// ---------------- degree (segment_sum of edge values + self loops) ----------
__global__ void degree_kernel(const int* __restrict__ row,
                              const float* __restrict__ val,
                              float* __restrict__ deg, int E, int N) {
    int i = blockIdx.x * blockDim.x + threadIdx.x;
    if (i >= E + N) return;
    if (i < E) {
        unsafeAtomicAdd(&deg[row[i]], val[i]);   // global_atomic_add_f32
    } else {
        unsafeAtomicAdd(&deg[i - E], 1.0f);      // self loop weight 1
    }
}

__global__ void dinv_kernel(const float* __restrict__ deg,
                            float* __restrict__ dinv, int N) {
    int i = blockIdx.x * blockDim.x + threadIdx.x;
    if (i >= N) return;
    float d = deg[i];
    dinv[i] = (d > 0.0f) ? rsqrtf(fmaxf(d, 1e-12f)) : 0.0f;
}

// ---------------- SpMM: Xout[r] += w * Xin[c], one wave32 per edge ----------
__global__ void spmm_kernel(const int* __restrict__ row,
                            const int* __restrict__ col,
                            const float* __restrict__ val,
                            const float* __restrict__ dinv,
                            const float* __restrict__ Xin,
                            float* __restrict__ Xout,
                            int E, int N) {
    int tid  = blockIdx.x * blockDim.x + threadIdx.x;
    int lane = tid & 31;
    int edge = tid >> 5;
    if (edge >= E + N) return;

    int r, c; float v;
    if (edge < E) {
        r = row[edge]; c = col[edge]; v = val[edge];
    } else {
        r = c = edge - E; v = 1.0f;
    }
    float w = dinv[r] * v * dinv[c];

    // 32 lanes x float4 = 512B contiguous gather of X[c]
    const float4* src = (const float4*)(Xin + (size_t)c * D_FEAT);
    float4 x = src[lane];

    float* dst = Xout + (size_t)r * D_FEAT + lane * 4;
    unsafeAtomicAdd(dst + 0, w * x.x);
    unsafeAtomicAdd(dst + 1, w * x.y);
    unsafeAtomicAdd(dst + 2, w * x.z);
    unsafeAtomicAdd(dst + 3, w * x.w);
}

// ---------------- logits = X2 @ lin via V_WMMA_F32_16X16X4_F32 --------------
// one wave (32 threads) per 16x16 output tile; K=128 in steps of 4.
// B comes from the transposed zero-padded Wpad[48][128] -> no divergent loads.
__global__ __launch_bounds__(32)
void gemm_wmma_kernel(const float* __restrict__ X,
                      const float* __restrict__ Wpad,
                      float* __restrict__ out, int N) {
    const int mbase = blockIdx.x * 16;
    const int nbase = blockIdx.y * 16;
    const int lane  = threadIdx.x;      // 0..31, EXEC all ones
    const int half  = lane >> 4;        // 0: K={0,1}, 1: K={2,3}
    const int mn    = lane & 15;

    const int mrow  = mbase + mn;
    const int mclmp = (mrow < N) ? mrow : (N - 1);
    const int ncol  = nbase + mn;       // < 48 always

    const float* ap = X    + (size_t)mclmp * D_FEAT + 2 * half;  // A row
    const float* bp = Wpad + (size_t)ncol  * D_FEAT + 2 * half;  // B col (transposed)

    v8f acc = {};
    #pragma unroll 8
    for (int k = 0; k < D_FEAT; k += 4) {
        // A: 16x4 f32 fragment (lanes 0-15: K=k,k+1; lanes 16-31: K=k+2,k+3)
        v2f a; a.x = ap[k];     a.y = ap[k + 1];
        // B: 4x16 f32 fragment, same K striping, contiguous in Wpad
        v2f b; b.x = bp[k];     b.y = bp[k + 1];

        acc = __builtin_amdgcn_wmma_f32_16x16x4_f32(
            /*neg_a=*/false, a, /*neg_b=*/false, b,
            /*c_mod=*/(short)0, acc, /*reuse_a=*/false, /*reuse_b=*/false);
    }

    // epilogue: C/D layout is VGPR j -> (M = mbase + j + 8*half, N = ncol)
    if (ncol < C_OUT) {
        float* op = out + (size_t)(mbase + 8 * half) * C_OUT + ncol;
        if (mbase + 16 <= N) {
            // wave-uniform full-tile fast path: no per-element guards
            #pragma unroll
            for (int j = 0; j < 8; ++j)
                op[(size_t)j * C_OUT] = acc[j];
        } else {
            #pragma unroll
            for (int j = 0; j < 8; ++j) {
                int m = mbase + j + 8 * half;
                if (m < N) op[(size_t)j * C_OUT] = acc[j];
            }
        }
    }
}

// ---------------- row-wise log_softmax over 40 logits, one wave per row -----
__global__ void logsoftmax_kernel(const float* __restrict__ logits,
                                  float* __restrict__ out, int N) {
    int tid  = blockIdx.x * blockDim.x + threadIdx.x;
    int lane = tid & 31;
    int r    = tid >> 5;
    if (r >= N) return;

    const float* Lr = logits + (size_t)r * C_OUT;
    float a = Lr[lane];                          // lanes 0..31 always valid
    bool bv = (lane + 32) < C_OUT;               // lanes 0..7
    float b = bv ? Lr[lane + 32] : -INFINITY;

    float m = fmaxf(a, b);
    #pragma unroll
    for (int off = 16; off > 0; off >>= 1)
        m = fmaxf(m, __shfl_xor(m, off, 32));

    float s = expf(a - m) + (bv ? expf(b - m) : 0.0f);
    #pragma unroll
    for (int off = 16; off > 0; off >>= 1)
        s += __shfl_xor(s, off, 32);

    float ls = logf(s);
    float* Or = out + (size_t)r * C_OUT;
    Or[lane] = a - m - ls;
    if (bv) Or[lane + 32] = b - m - ls;
}

// ---------------- launch ----------------
extern "C" void kernel_launch(void* const* d_in, const int* in_sizes, int n_in,
                              void* d_out, int out_size, void* d_ws, size_t ws_size,
                              hipStream_t stream) {
    const float* features = (const float*)d_in[0];
    const int*   adj_row  = (const int*)d_in[1];
    const int*   adj_col  = (const int*)d_in[2];
    const float* adj_val  = (const float*)d_in[3];
    const float* lin      = (const float*)d_in[4];

    const int N = in_sizes[0] / D_FEAT;   // 50000
    const int E = in_sizes[1];            // 800000
    const int M = E + N;                  // edges + self loops

    float* out    = (float*)d_out;
    float* logp   = out;                              // [N, 40]
    float* logits = logp + (size_t)N * C_OUT;         // [N, 40]
    float* X1     = logits + (size_t)N * C_OUT;       // [N, 128]
    float* X2     = X1 + (size_t)N * D_FEAT;          // [N, 128]

    float* deg  = (float*)d_ws;           // [N]
    float* dinv = deg + N;                // [N]
    float* Wpad = dinv + N;               // [48 * 128]

    const int ND = N * D_FEAT;

    // fresh accumulators every call (graph replays re-enter here)
    zero_f32<<<(N + 255) / 256, 256, 0, stream>>>(deg, N);
    zero_f32<<<(ND + 255) / 256, 256, 0, stream>>>(X1, ND);
    zero_f32<<<(ND + 255) / 256, 256, 0, stream>>>(X2, ND);
    pad_weights_kernel<<<(C_PAD * D_FEAT + 255) / 256, 256, 0, stream>>>(lin, Wpad);

    degree_kernel<<<(M + 255) / 256, 256, 0, stream>>>(adj_row, adj_val, deg, E, N);
    dinv_kernel<<<(N + 255) / 256, 256, 0, stream>>>(deg, dinv, N);

    const int spmm_blocks = (M * 32 + 255) / 256;     // one wave per edge
    spmm_kernel<<<spmm_blocks, 256, 0, stream>>>(adj_row, adj_col, adj_val,
                                                 dinv, features, X1, E, N);
    spmm_kernel<<<spmm_blocks, 256, 0, stream>>>(adj_row, adj_col, adj_val,
                                                 dinv, X1, X2, E, N);

    dim3 ggrid((N + 15) / 16, (C_OUT + 15) / 16);     // 3125 x 3 tiles
    gemm_wmma_kernel<<<ggrid, 32, 0, stream>>>(X2, Wpad, logits, N);

    logsoftmax_kernel<<<(N * 32 + 255) / 256, 256, 0, stream>>>(logits, logp, N);
}